// ProjectionMatcher_33758442947405
// MI455X (gfx1250) — compile-verified
//
#include <hip/hip_runtime.h>
#include <hip/hip_bf16.h>
#include <math.h>

typedef float f32x2 __attribute__((ext_vector_type(2)));
typedef float f32x8 __attribute__((ext_vector_type(8)));

#define DD 128
#define HW 65          // 128/2 + 1
#define GG 125
#define PSTR 66        // padded LDS row stride (col 65 zeroed)
#define PI_OVER_64 0.04908738521234052f   // == 2*pi/128

// ---------------------------------------------------------------------------
// V0: shift + sinc^2 weight the volume
// ---------------------------------------------------------------------------
__global__ __launch_bounds__(256) void prep_vol(const float* __restrict__ vol,
                                                float* __restrict__ wvol) {
  int o = blockIdx.x * 256 + threadIdx.x;
  if (o >= DD * DD * DD) return;
  int x = o & 127, y = (o >> 7) & 127, z = o >> 14;
  int sx = (x + 64) & 127, sy = (y + 64) & 127, sz = (z + 64) & 127;
  float fx = (float)(sx - 64) * (1.0f / 128.0f);
  float fy = (float)(sy - 64) * (1.0f / 128.0f);
  float fz = (float)(sz - 64) * (1.0f / 128.0f);
  float r = sqrtf(fx * fx + fy * fy + fz * fz);
  float xr = 3.14159265358979f * r;
  float s = (xr < 1e-6f) ? 1.0f : (sinf(xr) / xr);
  wvol[o] = vol[(sz * DD + sy) * DD + sx] * s * s;
}

// ---------------------------------------------------------------------------
// real -> complex DFT along fastest axis (length 128 -> 65 bins)
// ---------------------------------------------------------------------------
__global__ __launch_bounds__(256) void dft_x_real(const float* __restrict__ in,
                                                  float2* __restrict__ out, int total) {
  __shared__ float twc[128], tws[128];
  if (threadIdx.x < 128) {
    float s, c;
    sincosf((float)threadIdx.x * PI_OVER_64, &s, &c);
    twc[threadIdx.x] = c; tws[threadIdx.x] = s;
  }
  __syncthreads();
  int o = blockIdx.x * 256 + threadIdx.x;
  if (o >= total) return;
  int kx = o % HW; int row = o / HW;
  const float* src = in + row * DD;
  float ar = 0.f, ai = 0.f;
  for (int x = 0; x < DD; ++x) {
    float v = src[x];
    int p = (kx * x) & 127;
    ar += v * twc[p];
    ai -= v * tws[p];
  }
  out[o] = make_float2(ar, ai);
}

// real -> complex DFT along x of images, with fftshift folded into reads
__global__ __launch_bounds__(256) void dft_x_img(const float* __restrict__ imgs,
                                                 float2* __restrict__ out, int total) {
  __shared__ float twc[128], tws[128];
  if (threadIdx.x < 128) {
    float s, c;
    sincosf((float)threadIdx.x * PI_OVER_64, &s, &c);
    twc[threadIdx.x] = c; tws[threadIdx.x] = s;
  }
  __syncthreads();
  int o = blockIdx.x * 256 + threadIdx.x;
  if (o >= total) return;
  int kx = o % HW; int y = (o / HW) & 127; int b = o / (HW * DD);
  const float* src = imgs + b * DD * DD + ((y + 64) & 127) * DD;
  float ar = 0.f, ai = 0.f;
  for (int x = 0; x < DD; ++x) {
    float v = src[(x + 64) & 127];
    int p = (kx * x) & 127;
    ar += v * twc[p];
    ai -= v * tws[p];
  }
  out[(b * DD + y) * HW + kx] = make_float2(ar, ai);
}

// complex DFT along middle axis (stride HW), output fftshifted
__global__ __launch_bounds__(256) void dft_y_shift(const float2* __restrict__ in,
                                                   float2* __restrict__ out, int nouter) {
  __shared__ float twc[128], tws[128];
  if (threadIdx.x < 128) {
    float s, c;
    sincosf((float)threadIdx.x * PI_OVER_64, &s, &c);
    twc[threadIdx.x] = c; tws[threadIdx.x] = s;
  }
  __syncthreads();
  int o = blockIdx.x * 256 + threadIdx.x;
  int total = nouter * DD * HW;
  if (o >= total) return;
  int kx = o % HW; int h = (o / HW) & 127; int q = o / (HW * DD);
  int ky = (h + 64) & 127;
  const float2* src = in + (q * DD) * HW + kx;
  float ar = 0.f, ai = 0.f;
  for (int y = 0; y < DD; ++y) {
    float2 cv = src[y * HW];
    int p = (ky * y) & 127;
    float wr = twc[p], wi = -tws[p];
    ar += cv.x * wr - cv.y * wi;
    ai += cv.x * wi + cv.y * wr;
  }
  out[(q * DD + h) * HW + kx] = make_float2(ar, ai);
}

// complex DFT along outermost axis (stride 128*HW), output fftshifted
__global__ __launch_bounds__(256) void dft_z_shift(const float2* __restrict__ in,
                                                   float2* __restrict__ out) {
  __shared__ float twc[128], tws[128];
  if (threadIdx.x < 128) {
    float s, c;
    sincosf((float)threadIdx.x * PI_OVER_64, &s, &c);
    twc[threadIdx.x] = c; tws[threadIdx.x] = s;
  }
  __syncthreads();
  int o = blockIdx.x * 256 + threadIdx.x;
  if (o >= DD * DD * HW) return;
  int kx = o % HW; int h = (o / HW) & 127; int hz = o / (HW * DD);
  int kz = (hz + 64) & 127;
  const float2* src = in + h * HW + kx;
  float ar = 0.f, ai = 0.f;
  for (int z = 0; z < DD; ++z) {
    float2 cv = src[z * DD * HW];
    int p = (kz * z) & 127;
    float wr = twc[p], wi = -tws[p];
    ar += cv.x * wr - cv.y * wi;
    ai += cv.x * wi + cv.y * wr;
  }
  out[(hz * DD + h) * HW + kx] = make_float2(ar, ai);
}

// ---------------------------------------------------------------------------
// Fused per-(b,t,g) kernel: slice-extract + CTF + conj-product + cropped
// inverse DFT via f32 WMMA + windowed max reduction.
// ---------------------------------------------------------------------------
__global__ __launch_bounds__(256) void match_kernel(
    const float* __restrict__ ctf, const float* __restrict__ rotmats,
    const float* __restrict__ gridmats, const float2* __restrict__ volF,
    const float2* __restrict__ fimg, float* __restrict__ corrOut,
    int* __restrict__ shiftOut) {
  __shared__ float sRz[9];
  __shared__ float sTwc[128];
  __shared__ float sTws[128];
  __shared__ float sPR[64 * PSTR];
  __shared__ float sPI[64 * PSTR];
  __shared__ float sM1R[32 * PSTR];
  __shared__ float sM1I[32 * PSTR];
  __shared__ float redV[256];
  __shared__ int   redI[256];

  const int tid = threadIdx.x;
  const int btg = blockIdx.x;
  const int g = btg % GG;
  const int bt = btg / GG;
  const int b = bt >> 1;

  if (tid < 128) {   // exact twiddles: all phases are multiples of 2*pi/128
    float s, c;
    sincosf((float)tid * PI_OVER_64, &s, &c);
    sTwc[tid] = c; sTws[tid] = s;
  }
  if (tid == 0) {
    // expR = rot(b,t) @ grid(g);  Rzyx = expR[::-1, ::-1]
    const float* R = rotmats + bt * 9;
    const float* Gm = gridmats + g * 9;
    float e[9];
    #pragma unroll
    for (int i = 0; i < 3; ++i)
      #pragma unroll
      for (int k = 0; k < 3; ++k) {
        float s = 0.f;
        #pragma unroll
        for (int j = 0; j < 3; ++j) s += R[i * 3 + j] * Gm[j * 3 + k];
        e[i * 3 + k] = s;
      }
    #pragma unroll
    for (int i = 0; i < 3; ++i)
      #pragma unroll
      for (int j = 0; j < 3; ++j) sRz[i * 3 + j] = e[(2 - i) * 3 + (2 - j)];
  }
  __syncthreads();

  const float r01 = sRz[1], r02 = sRz[2];
  const float r11 = sRz[4], r12 = sRz[5];
  const float r21 = sRz[7], r22 = sRz[8];
  const float*  cb = ctf  + b * (DD * HW);
  const float2* fb = fimg + b * (DD * HW);
  __builtin_prefetch(cb, 0, 0);          // global_prefetch_b8
  __builtin_prefetch(fb, 0, 0);

  const int wave = tid >> 5;
  const int lane = tid & 31;
  const int mrow = lane & 15;
  const int basek = (lane >> 4) << 1;    // K split per A/B layout

  // Named accumulators (no register arrays -> no v_movrel shuffling).
  f32x8 accR0 = {0,0,0,0,0,0,0,0}, accI0 = {0,0,0,0,0,0,0,0};
  f32x8 accR1 = {0,0,0,0,0,0,0,0}, accI1 = {0,0,0,0,0,0,0,0};

  // Stage-1 tile body: M1 tile += Wy tile * P half.  Branchless row masking.
  auto stage1 = [&](int tt, int pass, f32x8& cR, f32x8& cI) {
    const int mt = tt / 5, nt = tt % 5;
    const int rr = mt * 16 + mrow;                 // Wy row (valid < 24)
    const int yr = 116 + rr;                       // spatial row after ifftshift
    const float vm = (rr < 24) ? 1.0f : 0.0f;      // branchless validity
    const int cc = nt * 16 + mrow;                 // P column
    const int ccc = (cc < PSTR) ? cc : (PSTR - 1); // clamped addr (col 65 is 0)
    const float cm = (cc < PSTR) ? 1.0f : 0.0f;
    #pragma unroll 2
    for (int kk = pass * 16; kk < pass * 16 + 16; ++kk) {
      int k0 = kk * 4 + basek;
      int k1 = k0 + 1;
      int p0 = (k0 * yr) & 127;
      int p1 = (k1 * yr) & 127;
      f32x2 aR, aI, aIn;
      aR.x = vm * sTwc[p0];  aR.y = vm * sTwc[p1];
      aI.x = vm * sTws[p0];  aI.y = vm * sTws[p1];
      aIn.x = -aI.x;         aIn.y = -aI.y;
      int lr0 = k0 & 63, lr1 = k1 & 63;
      f32x2 bR, bI;
      bR.x = cm * sPR[lr0 * PSTR + ccc];
      bR.y = cm * sPR[lr1 * PSTR + ccc];
      bI.x = cm * sPI[lr0 * PSTR + ccc];
      bI.y = cm * sPI[lr1 * PSTR + ccc];
      // M1R += WyR*PR - WyI*PI ;  M1I += WyR*PI + WyI*PR
      cR = __builtin_amdgcn_wmma_f32_16x16x4_f32(false, aR,  false, bR, (short)0, cR, false, false);
      cR = __builtin_amdgcn_wmma_f32_16x16x4_f32(false, aIn, false, bI, (short)0, cR, false, false);
      cI = __builtin_amdgcn_wmma_f32_16x16x4_f32(false, aR,  false, bI, (short)0, cI, false, false);
      cI = __builtin_amdgcn_wmma_f32_16x16x4_f32(false, aI,  false, bR, (short)0, cI, false, false);
    }
  };

  // K(=ky) split into two 64-row halves to keep LDS < 64KB.
  #pragma unroll 1
  for (int pass = 0; pass < 2; ++pass) {
    // ---- fill half of P: rows ky = pass*64 .. pass*64+63 (local row = hh) ----
    for (int idx = tid; idx < 64 * HW; idx += 256) {
      int hh = idx / HW, w = idx % HW;
      int h = pass ? hh : (hh + 64);         // ky=(h+64)&127, local row = hh
      float gy = (float)(h - 64) * (1.0f / 128.0f);
      float gx = (float)w * (1.0f / 128.0f);
      float rz = r01 * gy + r02 * gx;
      float ry = r11 * gy + r12 * gx;
      float rx = r21 * gy + r22 * gx;
      bool cj = rx < 0.f;
      if (cj) { rz = -rz; ry = -ry; rx = -rx; }
      float zi = rz * 128.f + 64.f;
      float yi = ry * 128.f + 64.f;
      float xi = rx * 128.f;
      float z0f = floorf(zi), y0f = floorf(yi), x0f = floorf(xi);
      float fz = zi - z0f, fy = yi - y0f, fx = xi - x0f;
      int z0 = (int)z0f, y0 = (int)y0f, x0 = (int)x0f;
      float sr = 0.f, si = 0.f;
      #pragma unroll
      for (int c = 0; c < 8; ++c) {
        int dz = c & 1, dy = (c >> 1) & 1, dx = (c >> 2) & 1;
        int zz = z0 + dz, yy = y0 + dy, xx = x0 + dx;
        float wgt = (dz ? fz : 1.f - fz) * (dy ? fy : 1.f - fy) * (dx ? fx : 1.f - fx);
        if (zz >= 0 && zz < DD && yy >= 0 && yy < DD && xx >= 0 && xx < HW) {
          float2 v = volF[(zz * DD + yy) * HW + xx];
          sr += wgt * v.x;
          si += wgt * v.y;
        }
      }
      if (cj) si = -si;
      float cv = cb[h * HW + w];
      float2 fv = fb[h * HW + w];
      // prod = fimg * conj(ctf * slice)
      sPR[hh * PSTR + w] = cv * (fv.x * sr + fv.y * si);
      sPI[hh * PSTR + w] = cv * (fv.y * sr - fv.x * si);
    }
    for (int r = tid; r < 64; r += 256) {  // zero pad column 65
      sPR[r * PSTR + HW] = 0.f;
      sPI[r * PSTR + HW] = 0.f;
    }
    __syncthreads();

    // ---- stage 1 partial over this ky half: 10 tiles on 8 waves ----
    stage1(wave, pass, accR0, accI0);
    if (wave < 2) stage1(wave + 8, pass, accR1, accI1);
    __syncthreads();
  }

  // ---- spill M1 tiles to LDS (D layout: VGPR v -> M = v + 8*(lane>=16)) ----
  {
    const int mb = (lane >> 4) << 3;
    {
      int mt = wave / 5, nt = wave % 5;
      int n = nt * 16 + mrow;
      if (n < PSTR) {
        #pragma unroll
        for (int v = 0; v < 8; ++v) {
          int m = mt * 16 + v + mb;
          sM1R[m * PSTR + n] = accR0[v];
          sM1I[m * PSTR + n] = accI0[v];
        }
      }
    }
    if (wave < 2) {
      int tt = wave + 8;
      int mt = tt / 5, nt = tt % 5;
      int n = nt * 16 + mrow;
      if (n < PSTR) {
        #pragma unroll
        for (int v = 0; v < 8; ++v) {
          int m = mt * 16 + v + mb;
          sM1R[m * PSTR + n] = accR1[v];
          sM1I[m * PSTR + n] = accI1[v];
        }
      }
    }
  }
  __syncthreads();

  // ---- stage 2: C(24x24) = Re(M1 * Wx), waves 0..3, 1 tile each ----
  f32x8 acc2 = {0, 0, 0, 0, 0, 0, 0, 0};
  const int mt2 = wave >> 1, nt2 = wave & 1;
  if (wave < 4) {
    const int arow = mt2 * 16 + mrow;
    const int cc = nt2 * 16 + mrow;
    const int xc = 116 + cc;
    const float S1 = 1.0f / 16384.0f;      // 1/D^2, alpha=1 (kx=0,64)
    const float S2 = 2.0f / 16384.0f;      // alpha=2 (hermitian fold)
    #pragma unroll 2
    for (int kk = 0; kk < 17; ++kk) {
      int k0 = kk * 4 + basek;
      int k1 = k0 + 1;
      int q0 = (k0 < PSTR) ? k0 : (PSTR - 1);   // clamped addr
      int q1 = (k1 < PSTR) ? k1 : (PSTR - 1);
      float m0f = (k0 < PSTR) ? 1.0f : 0.0f;
      float m1f = (k1 < PSTR) ? 1.0f : 0.0f;
      f32x2 aR, aI, bR, bIn;
      aR.x = m0f * sM1R[arow * PSTR + q0];
      aR.y = m1f * sM1R[arow * PSTR + q1];
      aI.x = m0f * sM1I[arow * PSTR + q0];
      aI.y = m1f * sM1I[arow * PSTR + q1];
      float a0 = (k0 == 0 || k0 == 64) ? S1 : (k0 < 65 ? S2 : 0.f);
      float a1 = (k1 < 65) ? S2 : 0.f;     // k1 is odd, never 0/64
      int p0 = (k0 * xc) & 127;
      int p1 = (k1 * xc) & 127;
      bR.x = a0 * sTwc[p0];   bR.y = a1 * sTwc[p1];
      bIn.x = -a0 * sTws[p0]; bIn.y = -a1 * sTws[p1];
      // C += M1R*WxR - M1I*WxI
      acc2 = __builtin_amdgcn_wmma_f32_16x16x4_f32(false, aR, false, bR,  (short)0, acc2, false, false);
      acc2 = __builtin_amdgcn_wmma_f32_16x16x4_f32(false, aI, false, bIn, (short)0, acc2, false, false);
    }
  }

  // ---- max + argmax over the 24x24 window ----
  float bv = -3.402823466e38f;
  int bi = 1 << 30;
  if (wave < 4) {
    int mb = (lane >> 4) << 3;
    int n = nt2 * 16 + mrow;
    #pragma unroll
    for (int v = 0; v < 8; ++v) {
      int m = mt2 * 16 + v + mb;
      if (m < 24 && n < 24) {
        float val = acc2[v];
        int id = m * 24 + n;
        if (val > bv || (val == bv && id < bi)) { bv = val; bi = id; }
      }
    }
  }
  redV[tid] = bv;
  redI[tid] = bi;
  __syncthreads();
  for (int s = 128; s >= 1; s >>= 1) {
    if (tid < s) {
      float ov = redV[tid + s]; int oi = redI[tid + s];
      if (ov > redV[tid] || (ov == redV[tid] && oi < redI[tid])) {
        redV[tid] = ov; redI[tid] = oi;
      }
    }
    __syncthreads();
  }
  if (tid == 0) {
    corrOut[btg] = redV[0];
    int m = redI[0] / 24, n = redI[0] % 24;
    shiftOut[btg * 2 + 0] = 52 + n;   // x
    shiftOut[btg * 2 + 1] = 52 + m;   // y
  }
}

// ---------------------------------------------------------------------------
// Final: per-batch top-2, mean/std (ddof=1), erf weight, gathers.
// ---------------------------------------------------------------------------
__global__ __launch_bounds__(32) void final_kernel(const float* __restrict__ corr,
                                                   const int* __restrict__ shifts,
                                                   const float* __restrict__ gridmats,
                                                   float* __restrict__ out) {
  int b = threadIdx.x;
  if (b >= 4) return;
  const float* c = corr + b * 250;
  float best1 = -3.402823466e38f, best2 = -3.402823466e38f;
  int i1 = 0, i2 = 0;
  float sum = 0.f, sumsq = 0.f;
  for (int j = 0; j < 250; ++j) {
    float v = c[j];
    sum += v;
    sumsq += v * v;
    if (v > best1) { best2 = best1; i2 = i1; best1 = v; i1 = j; }
    else if (v > best2) { best2 = v; i2 = j; }
  }
  float mean = sum * (1.0f / 250.0f);
  float var = (sumsq - 250.0f * mean * mean) * (1.0f / 249.0f);
  float sd = sqrtf(fmaxf(var, 0.f));
  float vals[2] = {best1, best2};
  int idxs[2] = {i1, i2};
  #pragma unroll
  for (int k = 0; k < 2; ++k) {
    int fi = idxs[k];
    int g = fi % GG;
    out[b * 2 + k] = vals[k];                                    // maxCorrs (4,2)
    const float* Gm = gridmats + g * 9;
    #pragma unroll
    for (int e = 0; e < 9; ++e) out[8 + (b * 2 + k) * 9 + e] = Gm[e];  // predRotMats
    int sx = shifts[(b * 250 + fi) * 2 + 0];
    int sy = shifts[(b * 250 + fi) * 2 + 1];
    out[80 + (b * 2 + k) * 2 + 0] = -((float)sx - 64.f) * 1.5f;  // predShiftsAngs
    out[80 + (b * 2 + k) * 2 + 1] = -((float)sy - 64.f) * 1.5f;
    float w = 0.5f * (1.f + erff((vals[k] - mean) / (sd * 1.41421356237f)));
    out[96 + b * 2 + k] = w;                                     // comparedWeight
  }
}

// ---------------------------------------------------------------------------
extern "C" void kernel_launch(void* const* d_in, const int* in_sizes, int n_in,
                              void* d_out, int out_size, void* d_ws, size_t ws_size,
                              hipStream_t stream) {
  const float* vol  = (const float*)d_in[0];   // (128,128,128)
  const float* imgs = (const float*)d_in[1];   // (4,128,128)
  const float* ctf  = (const float*)d_in[2];   // (4,128,65)
  const float* rotm = (const float*)d_in[3];   // (4,2,3,3)
  const float* grid = (const float*)d_in[4];   // (125,3,3)

  const size_t NC = (size_t)DD * DD * HW;      // 1,064,960 complex (volume rfft)
  const size_t NF = (size_t)4 * DD * HW;       // 33,280 complex (image rfft)
  float* ws = (float*)d_ws;
  float*  regA   = ws;                         // t1 then volF       (2*NC floats)
  float*  regB   = ws + 2 * NC;                // wvol then t2       (2*NC floats)
  float2* ftmp   = (float2*)(ws + 4 * NC);     // image x-DFT        (NF cplx)
  float2* fimgws = (float2*)(ws + 4 * NC + 2 * NF);
  float*  corrws = ws + 4 * NC + 4 * NF;       // 1000 floats
  int*    shifts = (int*)(corrws + 1000);      // 2000 ints

  // volume rfft: weight/shift, then separable DFTs with fftshift folded in
  prep_vol<<<(DD * DD * DD) / 256, 256, 0, stream>>>(vol, regB);
  dft_x_real<<<(int)((NC + 255) / 256), 256, 0, stream>>>(regB, (float2*)regA, (int)NC);
  dft_y_shift<<<(int)((NC + 255) / 256), 256, 0, stream>>>((float2*)regA, (float2*)regB, DD);
  dft_z_shift<<<(int)((NC + 255) / 256), 256, 0, stream>>>((float2*)regB, (float2*)regA);

  // image spectra
  dft_x_img<<<(int)((NF + 255) / 256), 256, 0, stream>>>(imgs, ftmp, (int)NF);
  dft_y_shift<<<(int)((NF + 255) / 256), 256, 0, stream>>>(ftmp, fimgws, 4);

  // fused match: 1000 candidates, one WG each (WMMA cropped inverse DFT)
  match_kernel<<<4 * 2 * GG, 256, 0, stream>>>(ctf, rotm, grid, (float2*)regA,
                                               fimgws, corrws, shifts);

  // per-batch top-2 + stats
  final_kernel<<<1, 32, 0, stream>>>(corrws, shifts, grid, (float*)d_out);
}